// MultimodalFusionModule_74929999446262
// MI455X (gfx1250) — compile-verified
//
#include <hip/hip_runtime.h>
#include <hip/hip_bf16.h>

// Problem dimensions (fixed by the reference setup_inputs()).
constexpr int B      = 64;
constexpr int T_VIS  = 256;
constexpr int D_VIS  = 512;
constexpr int T_PROP = 1024;
constexpr int D_PROP = 128;
constexpr int L_LANG = 77;
constexpr int D_LANG = 512;
constexpr int TT     = 512;
constexpr int D_OUT  = D_VIS + D_PROP + D_LANG;   // 1152

typedef __attribute__((ext_vector_type(2))) float v2f;
typedef __attribute__((ext_vector_type(8))) float v8f;

// ---------------------------------------------------------------------------
// Kernel 1: lang mean via exact f32 WMMA column-sum.
//   mean[b, d] = (1/77) * sum_l lang[b, l, d]
// One wave handles one (batch, 16-column tile). A = ones(16x4) so
// D[m,n] = sum_k B[k,n]; accumulate over K=77 in chunks of 4 (tail masked by
// zeroing B so EXEC stays all-1s, as WMMA requires).
// B-matrix f32 4x16 layout (2 VGPRs/lane): lanes 0-15 hold K=0,1 (N=lane),
// lanes 16-31 hold K=2,3 (N=lane-16). A layout is irrelevant (all ones).
// ---------------------------------------------------------------------------
__global__ void __launch_bounds__(256)
lang_mean_wmma_kernel(const float* __restrict__ lang, float* __restrict__ lmean) {
    const int wave  = blockIdx.x * 8 + (threadIdx.x >> 5);  // global wave id
    const int b     = wave >> 5;                            // 32 tiles per batch
    const int d0    = (wave & 31) << 4;                     // column tile base
    const int lane  = threadIdx.x & 31;
    const int n     = lane & 15;
    const int khalf = lane >> 4;                            // 0: K={0,1}, 1: K={2,3}

    const float* base = lang + (size_t)b * L_LANG * D_LANG + d0 + n;

    v2f a; a.x = 1.0f; a.y = 1.0f;      // ones -> column sums
    v8f c = {};

    #pragma unroll 4
    for (int l0 = 0; l0 < L_LANG; l0 += 4) {
        const int k0 = l0 + khalf * 2;
        const int k1 = k0 + 1;
        // Mask out-of-range K by multiplying by 0 (no EXEC divergence).
        const float m0 = (k0 < L_LANG) ? 1.0f : 0.0f;
        const float m1 = (k1 < L_LANG) ? 1.0f : 0.0f;
        const int ck0 = k0 < L_LANG ? k0 : (L_LANG - 1);
        const int ck1 = k1 < L_LANG ? k1 : (L_LANG - 1);
        v2f bv;
        bv.x = base[(size_t)ck0 * D_LANG] * m0;
        bv.y = base[(size_t)ck1 * D_LANG] * m1;
        // D = A(16x4, ones) * B(4x16) + C   -> v_wmma_f32_16x16x4_f32
        c = __builtin_amdgcn_wmma_f32_16x16x4_f32(
                /*neg_a=*/false, a, /*neg_b=*/false, bv,
                /*c_mod=*/(short)0, c, /*reuse_a=*/false, /*reuse_b=*/false);
    }

    // Every row of D equals the column sum; C/D VGPR0: lanes 0-15 -> (M=0, N=lane).
    if (lane < 16) {
        lmean[(size_t)b * D_LANG + d0 + lane] = c[0] * (1.0f / (float)L_LANG);
    }
}

// ---------------------------------------------------------------------------
// Kernel 2: fused interpolation + concat (pure streaming, bandwidth-bound).
// One block per (b, t); 288 threads; one float4 of the 1152-float row each.
// ---------------------------------------------------------------------------
__device__ __forceinline__ int lower_bound_f(const float* __restrict__ ts, int n, float x) {
    int lo = 0, hi = n;
    while (lo < hi) {
        const int mid = (lo + hi) >> 1;
        if (ts[mid] < x) lo = mid + 1; else hi = mid;   // side='left'
    }
    return lo;
}

__global__ void __launch_bounds__(288)
fusion_kernel(const float* __restrict__ vis,  const float* __restrict__ vts,
              const float* __restrict__ prop, const float* __restrict__ pts,
              const float* __restrict__ lmean, const float* __restrict__ tgt,
              float* __restrict__ out) {
    const int bt  = blockIdx.x;            // 0 .. B*TT-1
    const int b   = bt / TT;
    const int tid = threadIdx.x;

    __shared__ int   s_vidx, s_pidx;
    __shared__ float s_vw,   s_pw;

    const float target = tgt[bt];

    if (tid == 0) {
        const float* ts = vts + (size_t)b * T_VIS;
        int i = lower_bound_f(ts, T_VIS, target);
        i = min(i, T_VIS - 2);
        const float tl = ts[i], tr = ts[i + 1];
        const float w = (target - tl) / (tr - tl + 1e-8f);
        s_vidx = i;
        s_vw   = fminf(fmaxf(w, 0.0f), 1.0f);
    } else if (tid == 32) {
        const float* ts = pts + (size_t)b * T_PROP;
        int i = lower_bound_f(ts, T_PROP, target);
        i = min(i, T_PROP - 2);
        const float tl = ts[i], tr = ts[i + 1];
        const float w = (target - tl) / (tr - tl + 1e-8f);
        s_pidx = i;
        s_pw   = fminf(fmaxf(w, 0.0f), 1.0f);
    }
    __syncthreads();

    float4* __restrict__ outrow = (float4*)(out + (size_t)bt * D_OUT);

    if (tid < D_VIS / 4) {                               // vision: float4 0..127
        const int   i = s_vidx;
        const float w = s_vw;
        const float4* fl = (const float4*)(vis + ((size_t)b * T_VIS + i)     * D_VIS);
        const float4* fr = (const float4*)(vis + ((size_t)b * T_VIS + i + 1) * D_VIS);
        const float4 L = fl[tid], R = fr[tid];
        float4 r;
        r.x = (1.0f - w) * L.x + w * R.x;
        r.y = (1.0f - w) * L.y + w * R.y;
        r.z = (1.0f - w) * L.z + w * R.z;
        r.w = (1.0f - w) * L.w + w * R.w;
        outrow[tid] = r;
    } else if (tid < (D_VIS + D_PROP) / 4) {             // proprio: float4 128..159
        const int   j = tid - D_VIS / 4;
        const int   i = s_pidx;
        const float w = s_pw;
        const float4* fl = (const float4*)(prop + ((size_t)b * T_PROP + i) * D_PROP);
        const float4* fr = fl + D_PROP / 4;
        const float4 L = fl[j], R = fr[j];
        float4 r;
        r.x = (1.0f - w) * L.x + w * R.x;
        r.y = (1.0f - w) * L.y + w * R.y;
        r.z = (1.0f - w) * L.z + w * R.z;
        r.w = (1.0f - w) * L.w + w * R.w;
        outrow[tid] = r;
    } else {                                             // lang broadcast: 160..287
        const int j = tid - (D_VIS + D_PROP) / 4;
        const float4* lm = (const float4*)(lmean + (size_t)b * D_LANG);
        outrow[tid] = lm[j];
    }
}

// ---------------------------------------------------------------------------
extern "C" void kernel_launch(void* const* d_in, const int* in_sizes, int n_in,
                              void* d_out, int out_size, void* d_ws, size_t ws_size,
                              hipStream_t stream) {
    (void)in_sizes; (void)n_in; (void)out_size; (void)ws_size;
    const float* vis  = (const float*)d_in[0];   // [B, T_VIS, D_VIS]
    const float* vts  = (const float*)d_in[1];   // [B, T_VIS]
    const float* prop = (const float*)d_in[2];   // [B, T_PROP, D_PROP]
    const float* pts  = (const float*)d_in[3];   // [B, T_PROP]
    const float* lang = (const float*)d_in[4];   // [B, L_LANG, D_LANG]
    const float* tgt  = (const float*)d_in[5];   // [B, TT]
    float* out   = (float*)d_out;                // [B, TT, D_OUT]
    float* lmean = (float*)d_ws;                 // [B, D_LANG] scratch (128 KB)

    // 1) lang mean via f32 WMMA column sums: 2048 waves = 256 blocks * 8 waves.
    lang_mean_wmma_kernel<<<(B * (D_LANG / 16)) / 8, 256, 0, stream>>>(lang, lmean);

    // 2) fused interpolate + concat: one block per output row.
    fusion_kernel<<<B * TT, 288, 0, stream>>>(vis, vts, prop, pts, lmean, tgt, out);
}